// STConv_68822555951228
// MI455X (gfx1250) — compile-verified
//
#include <hip/hip_runtime.h>
#include <hip/hip_bf16.h>

typedef __attribute__((ext_vector_type(2))) float        v2f;
typedef __attribute__((ext_vector_type(8))) float        v8f;
typedef __attribute__((ext_vector_type(4))) unsigned int v4u;
typedef __attribute__((ext_vector_type(8))) int          v8i;
typedef __attribute__((ext_vector_type(4))) int          v4i;

#define B_    16
#define CH_   32
#define CIN_  64
#define OUT_  64
#define H_    256
#define W_    256
#define H2_   128
#define W2_   128
#define K3_   192      // 3*CIN
#define TP_   64       // pixels per tile
#define STR_  68       // LDS row pitch (64 + 4 DW pad; 16B aligned, TDM-expressible)
#define CST_  132      // LDS col stride for stem input tile (130 + pad)

#if defined(__has_include)
#if __has_include(<hip/amd_detail/amd_gfx1250_TDM.h>)
#define TDM_6ARG 1
#endif
#endif

// -------------------------------------------------------------------------
// Kernel 1: stem conv 3x3 stride-2 pad-1 (32->32) + bias + SiLU, as implicit
// GEMM on WMMA f32 16x16x4.  M = 16 pixels, N = 16 out-channels, K = 288.
// K-order: kappa = tap*32 + ci  ->  per-WMMA the 4 K values are 4 consecutive
// ci with a fixed (ki,kj) tap, so fragment addresses need no runtime div/mod.
// -------------------------------------------------------------------------
__global__ __launch_bounds__(256)
void stconv_stem(const float* __restrict__ mas, const float* __restrict__ w_inp,
                 const float* __restrict__ b_inp, float* __restrict__ x)
{
    __shared__ float xt[CH_ * 3 * CST_];   // 32ch x 3 rows x 132 = 50688 B

    const int tid = threadIdx.x;
    const int b   = blockIdx.x >> 8;
    const int t   = blockIdx.x & 255;
    const int i   = t >> 1;               // output row
    const int j0  = (t & 1) * TP_;        // half-row start (64 out pixels)

    // ---- stage mas halo tile: rows 2i-1..2i+1, cols 2j0-1..2j0+128, 0-padded
    const int gi0 = 2 * i - 1;
    const int gj0 = 2 * j0 - 1;
    for (int idx = tid; idx < CH_ * 3 * CST_; idx += 256) {
        const int ci  = idx / (3 * CST_);
        const int rem = idx - ci * (3 * CST_);
        const int r   = rem / CST_;
        const int col = rem - r * CST_;
        const int gi  = gi0 + r;
        const int gj  = gj0 + col;
        float v = 0.f;
        if (col < 130 && gi >= 0 && gi < H_ && gj >= 0 && gj < W_)
            v = mas[(size_t)((b * CH_ + ci) * H_ + gi) * W_ + gj];
        xt[idx] = v;
    }
    __syncthreads();

    // ---- WMMA phase: 8 waves = 4 pixel-tiles x 2 channel-tiles
    const int lane = tid & 31;
    const int wave = tid >> 5;
    const int lm   = lane & 15;
    const int half = lane >> 4;           // lanes 16..31: K+2 / M+8
    const int M0   = (wave & 3) * 16;     // pixel offset in tile
    const int N0   = (wave >> 2) * 16;    // out-channel offset (0 or 16)

    const float* wrow = w_inp + (N0 + lm) * (CH_ * 9);  // [c_out][ci*9 + tap]
    const int colA = 2 * (M0 + lm);       // stride-2 pixel -> input col

    v8f acc = {};
#pragma unroll
    for (int tap = 0; tap < 9; ++tap) {
        const int ki = tap / 3;           // compile-time after unroll
        const int kj = tap % 3;
#pragma unroll
        for (int ci0 = 0; ci0 < CH_; ci0 += 4) {
            const int ci = ci0 + 2 * half;
            v2f a, bb;
            a.x  = xt[(ci * 3 + ki) * CST_ + colA + kj];         // A[lm][k]
            a.y  = xt[((ci + 1) * 3 + ki) * CST_ + colA + kj];   // A[lm][k+1]
            bb.x = wrow[ci * 9 + tap];                           // B[k][lm]
            bb.y = wrow[(ci + 1) * 9 + tap];                     // B[k+1][lm]
            acc = __builtin_amdgcn_wmma_f32_16x16x4_f32(
                      false, a, false, bb, (short)0, acc, false, false);
        }
    }

    // ---- epilogue: bias + SiLU, pixels contiguous in r -> float4 stores
    const float bias = b_inp[N0 + lm];
    float res[8];
#pragma unroll
    for (int r = 0; r < 8; ++r) {
        const float y = acc[r] + bias;
        res[r] = y / (1.f + __expf(-y));
    }
    float* xp = x + (size_t)((b * CH_ + N0 + lm) * H2_ + i) * W2_
                  + j0 + M0 + 8 * half;
    reinterpret_cast<float4*>(xp)[0] = make_float4(res[0], res[1], res[2], res[3]);
    reinterpret_cast<float4*>(xp)[1] = make_float4(res[4], res[5], res[6], res[7]);
}

// -------------------------------------------------------------------------
// Kernel 2: deltas + grouped 1x1 gates + concat + out_conv (WMMA f32 16x16x4)
// block: 256 threads (8 waves), one batch x 64 contiguous pixels (half row).
// The cen slab (rows 128..191 of the LDS outs tile) is fetched by the Tensor
// Data Mover: one 2D descriptor (tile 64x64 f32, global row stride 16384 el,
// LDS pitch 68 DW via pad_interval=64DW / pad_amount=4DW), overlapped with
// the phase-1 gate math and ordered by s_wait_tensorcnt + barrier.
// -------------------------------------------------------------------------
__global__ __launch_bounds__(256)
void stconv_fused(const float* __restrict__ x, const float* __restrict__ cen,
                  const float* __restrict__ w1, const float* __restrict__ w2,
                  const float* __restrict__ w3, const float* __restrict__ w_out,
                  const float* __restrict__ b_out, float* __restrict__ out)
{
    __shared__ float outs[K3_ * STR_];   // 192 x 68 floats = 52224 B
    __shared__ float wg[3 * CH_ * 16];   // w1|w2|w3 = 1536 floats = 6144 B

    const int tid = threadIdx.x;
    const int b   = blockIdx.x >> 8;
    const int t   = blockIdx.x & 255;
    const int i   = t >> 1;               // image row
    const int j0  = (t & 1) * TP_;        // half-row start
    const int pbase = i * W2_ + j0;       // flat pixel base of tile
    const int wave  = tid >> 5;

    // keep the 48 KB w_out matrix hot in cache for the WMMA phase
    __builtin_prefetch(w_out + (size_t)tid * 48, 0, 1);

#if __has_builtin(__builtin_amdgcn_tensor_load_to_lds)
    // ---- TDM: DMA cen tile -> outs rows 128..191 (one descriptor, wave 0)
    if (wave == 0) {
        const uint64_t ga = (uint64_t)(uintptr_t)
            (cen + (size_t)((b * CIN_) * H2_ + i) * W2_ + j0);
        const uint32_t lds = (uint32_t)(uintptr_t)outs + (2 * CIN_) * STR_ * 4;

        v4u g0;
        g0.x = 1u;                                   // count=1, user descriptor
        g0.y = lds;                                  // lds_addr (bytes)
        g0.z = (uint32_t)ga;                         // global_addr[31:0]
        g0.w = (uint32_t)(ga >> 32) | (2u << 30);    // global_addr[56:32] | type=2

        const uint32_t td0 = 64, td1 = 64;           // tensor dims (elements)
        const uint32_t tl0 = 64, tl1 = 64, tl2 = 0;  // tile dims
        const uint64_t s0  = H2_ * W2_;              // dim0 stride = 16384 el
        v8i g1;
        g1[0] = (int)((2u << 16)      // data_size = 4B
                    | (1u << 20)      // pad_enable
                    | (5u << 22)      // pad_interval: 64 DWORDs
                    | (3u << 25));    // pad_amount:   4 DWORDs
        g1[1] = (int)((td0 & 0xFFFFu) << 16);
        g1[2] = (int)((td0 >> 16) | ((td1 & 0xFFFFu) << 16));
        g1[3] = (int)((td1 >> 16) | (tl0 << 16));
        g1[4] = (int)(tl1 | (tl2 << 16));
        g1[5] = (int)(s0 & 0xFFFFFFFFu);
        g1[6] = (int)(s0 >> 32);
        g1[7] = 0;
        v4i g2 = {0, 0, 0, 0};
        v4i g3 = {0, 0, 0, 0};
#ifdef TDM_6ARG
        v8i g4 = {0, 0, 0, 0, 0, 0, 0, 0};
        __builtin_amdgcn_tensor_load_to_lds(g0, g1, g2, g3, g4, 0);
#else
        __builtin_amdgcn_tensor_load_to_lds(g0, g1, g2, g3, 0);
#endif
    }
#endif

    // stage the tiny grouped-conv weights
    for (int idx = tid; idx < 512; idx += 256) {
        wg[idx]        = w1[idx];
        wg[idx + 512]  = w2[idx];
        wg[idx + 1024] = w3[idx];
    }
    __syncthreads();

    // ---- Phase 1: deltas -> gates -> outs rows 0..127 (overlaps the TDM DMA)
    const int pl   = tid & 63;            // pixel within tile
    const int crow = tid >> 6;            // 0..3
    const int j    = j0 + pl;

    const int DY[8] = {-3, -3, -3, 0, 3, 3, 3, 0};
    const int DX[8] = {-3,  0,  3, 3, 3, 0, -3, -3};

#pragma unroll
    for (int it = 0; it < 8; ++it) {
        const int c = crow * 8 + it;
        const float* xb = x + (size_t)((b * CH_ + c) * H2_) * W2_;
        const float xc = xb[i * W2_ + j];
        float d[8];
#pragma unroll
        for (int k = 0; k < 8; ++k) {
            const int ii = i + DY[k];
            const int jj = j + DX[k];
            const float v = (ii >= 0 && ii < H2_ && jj >= 0 && jj < W2_)
                                ? xb[ii * W2_ + jj] : 0.f;
            d[k] = xc - v;
        }
        const float* W1c = wg + c * 16;
        const float* W2c = wg + 512 + c * 16;
        const float* W3c = wg + 1024 + c * 16;
#pragma unroll
        for (int o = 0; o < 2; ++o) {
            float g1v = 0.f, g2v = 0.f, g3v = 0.f;
#pragma unroll
            for (int k = 0; k < 8; ++k) {
                g1v = fmaf(d[k], W1c[o * 8 + k], g1v);
                g2v = fmaf(d[k], W2c[o * 8 + k], g2v);
                g3v = fmaf(d[k], W3c[o * 8 + k], g3v);
            }
            outs[(2 * c + o) * STR_ + pl]        = g1v * g2v;  // square
            outs[(CIN_ + 2 * c + o) * STR_ + pl] = g3v;        // dout
        }
    }

#if __has_builtin(__builtin_amdgcn_tensor_load_to_lds)
    __builtin_amdgcn_s_wait_tensorcnt(0);   // free for waves that issued nothing
#else
    // fallback: manual cen staging, rows 128..191
    for (int idx = tid; idx < CIN_ * TP_; idx += 256) {
        const int cc = idx >> 6;
        const int p  = idx & 63;
        outs[(2 * CIN_ + cc) * STR_ + p] =
            cen[(size_t)((b * CIN_ + cc) * H2_ + i) * W2_ + j0 + p];
    }
#endif
    __syncthreads();

    // ---- Phase 2: out[64 x tile] = w_out[64 x 192] * outs[192 x tile] + bias
    const int lane = tid & 31;
    const int lm   = lane & 15;
    const int half = lane >> 4;
    const int M0   = (wave & 3) * 16;     // pixel offset within tile
    const int nt0  = (wave >> 2) * 2;     // 0 or 2
    const int N0a  = nt0 * 16;
    const int N0b  = (nt0 + 1) * 16;

    v8f acc0 = {}; v8f acc1 = {};
    const float* wra = w_out + (N0a + lm) * K3_;   // row of w_out (L2-hot)
    const float* wrb = w_out + (N0b + lm) * K3_;

#pragma unroll 4
    for (int k0 = 0; k0 < K3_; k0 += 4) {
        const int ka = k0 + 2 * half;     // this lane-half's K pair
        v2f a, bA, bB;
        a.x  = outs[ka * STR_ + M0 + lm];          // A[M=lm][ka]
        a.y  = outs[(ka + 1) * STR_ + M0 + lm];    // A[M=lm][ka+1]
        bA.x = wra[ka];  bA.y = wra[ka + 1];       // B[ka..][N=lm], tile a
        bB.x = wrb[ka];  bB.y = wrb[ka + 1];       // B[ka..][N=lm], tile b
        acc0 = __builtin_amdgcn_wmma_f32_16x16x4_f32(
                   false, a, false, bA, (short)0, acc0, false, false);
        acc1 = __builtin_amdgcn_wmma_f32_16x16x4_f32(
                   false, a, false, bB, (short)0, acc1, false, false);
    }

    // D element (vgpr r, lane): pixel = M0 + r + 8*half, channel = N0 + lm
    const float biasA = b_out[N0a + lm];
    const float biasB = b_out[N0b + lm];
#pragma unroll
    for (int r = 0; r < 8; ++r) {
        const int pix = pbase + M0 + r + 8 * half;
        out[(size_t)(b * OUT_ + N0a + lm) * (H2_ * W2_) + pix] = acc0[r] + biasA;
        out[(size_t)(b * OUT_ + N0b + lm) * (H2_ * W2_) + pix] = acc1[r] + biasB;
    }
}

// -------------------------------------------------------------------------
extern "C" void kernel_launch(void* const* d_in, const int* in_sizes, int n_in,
                              void* d_out, int out_size, void* d_ws, size_t ws_size,
                              hipStream_t stream) {
    const float* cen   = (const float*)d_in[0];
    const float* mas   = (const float*)d_in[1];
    const float* w_inp = (const float*)d_in[2];
    const float* b_inp = (const float*)d_in[3];
    const float* w1    = (const float*)d_in[4];
    const float* w2    = (const float*)d_in[5];
    const float* w3    = (const float*)d_in[6];
    const float* w_out = (const float*)d_in[7];
    const float* b_out = (const float*)d_in[8];
    float* out = (float*)d_out;
    float* x   = (float*)d_ws;   // stem activations: 16*32*128*128 f32 = 33.5 MB

    stconv_stem<<<B_ * 256, 256, 0, stream>>>(mas, w_inp, b_inp, x);
    stconv_fused<<<B_ * 256, 256, 0, stream>>>(x, cen, w1, w2, w3,
                                               w_out, b_out, out);
}